// MOELayer_raw_3521873183219
// MI455X (gfx1250) — compile-verified
//
#include <hip/hip_runtime.h>
#include <hip/hip_bf16.h>

#define NUM_EXPERT 32
#define IN_FEAT 512
#define OUT_FEAT 512
#define BATCH 4096
#define TILE_M 16
#define MAX_TILES (BATCH / TILE_M + NUM_EXPERT) // 288 upper bound

typedef __attribute__((ext_vector_type(2))) float v2f;
typedef __attribute__((ext_vector_type(4))) float v4f;
typedef __attribute__((ext_vector_type(8))) float v8f;

// Explicit global-address-space pointers so hot loads/stores lower to
// global_load_*/global_store_* (LOADcnt/STOREcnt only) instead of flat_*
// (which also ties up DScnt and couples with the LDS pipeline).
typedef const float  __attribute__((address_space(1)))* gcfp;
typedef const v2f    __attribute__((address_space(1)))* gcv2fp;
typedef const v4f    __attribute__((address_space(1)))* gcv4fp;
typedef float        __attribute__((address_space(1)))* gfp;

struct MoeWs {
  int counts[NUM_EXPERT];
  int cursor[NUM_EXPERT];
  int offsets[NUM_EXPERT + 1];
  int numTiles;
  int tileExpert[MAX_TILES];
  int tileM[MAX_TILES];
  int perm[BATCH];
};

__global__ void moe_zero(MoeWs* ws) {
  int t = threadIdx.x;
  if (t < NUM_EXPERT) { ws->counts[t] = 0; ws->cursor[t] = 0; }
  if (t == 0) ws->numTiles = 0;
}

__global__ void moe_hist(const int* __restrict__ gate, MoeWs* ws) {
  int b = blockIdx.x * blockDim.x + threadIdx.x;
  if (b < BATCH) atomicAdd(&ws->counts[gate[b] & (NUM_EXPERT - 1)], 1);
}

__global__ void moe_scan(MoeWs* ws) {
  if (threadIdx.x != 0) return;
  int off = 0;
  for (int e = 0; e < NUM_EXPERT; ++e) { ws->offsets[e] = off; off += ws->counts[e]; }
  ws->offsets[NUM_EXPERT] = off;
  int nt = 0;
  for (int e = 0; e < NUM_EXPERT; ++e) {
    int tiles = (ws->counts[e] + TILE_M - 1) / TILE_M;
    for (int j = 0; j < tiles; ++j) { ws->tileExpert[nt] = e; ws->tileM[nt] = j; ++nt; }
  }
  ws->numTiles = nt;
}

__global__ void moe_scatter(const int* __restrict__ gate, MoeWs* ws) {
  int b = blockIdx.x * blockDim.x + threadIdx.x;
  if (b < BATCH) {
    int e = gate[b] & (NUM_EXPERT - 1);
    int pos = ws->offsets[e] + atomicAdd(&ws->cursor[e], 1);
    ws->perm[pos] = b;
  }
}

// One block = one (expert, 16-token tile) x 256 output columns.
// 128 threads = 4 waves; each wave owns 64 columns (4 WMMA n-tiles).
__launch_bounds__(128)
__global__ void moe_gemm(const float* __restrict__ inp,
                         const float* __restrict__ weight,
                         float* __restrict__ out,
                         const MoeWs* __restrict__ ws) {
  __shared__ __align__(16) float sA[TILE_M][IN_FEAT]; // 32 KB activations
  __shared__ int sTok[TILE_M];

  const int tile = blockIdx.x;
  if (tile >= ws->numTiles) return; // block-uniform exit
  const int e = ws->tileExpert[tile];
  const int mt = ws->tileM[tile];
  const int base = ws->offsets[e] + mt * TILE_M;
  const int mcount = min(TILE_M, ws->counts[e] - mt * TILE_M);

  if (threadIdx.x < TILE_M)
    sTok[threadIdx.x] = (threadIdx.x < mcount) ? ws->perm[base + threadIdx.x] : -1;
  __syncthreads();

  // Stage A tile: 16 rows x 512 f32 via global_load_b128 (zero-pad missing rows).
  for (int f = threadIdx.x; f < TILE_M * (IN_FEAT / 4); f += 128) {
    int row = f >> 7;      // 128 float4 per row
    int c4  = f & 127;
    int tok = sTok[row];
    v4f v = {};
    if (tok >= 0) v = ((gcv4fp)(inp + (size_t)tok * IN_FEAT))[c4];
    ((v4f*)&sA[row][0])[c4] = v;
  }
  __syncthreads();

  const int lane = threadIdx.x & 31;
  const int wave = threadIdx.x >> 5;
  const int hl   = lane >> 4;  // half-wave select
  const int l15  = lane & 15;

  const float* wbase = weight + (size_t)e * OUT_FEAT * IN_FEAT;
  const int ncol0 = blockIdx.y * 256 + wave * 64;

  // B fragment: lane holds column (ncol + l15), K rows 2*hl and 2*hl+1.
  gcfp bptr[4];
#pragma unroll
  for (int nt = 0; nt < 4; ++nt) {
    const float* p = wbase + (size_t)(ncol0 + nt * 16 + l15) * IN_FEAT + 2 * hl;
    __builtin_prefetch(p, 0, 1);
    bptr[nt] = (gcfp)p;
  }

  v8f acc[4] = {};
  // A fragment from LDS: lane holds row l15, K = k0 + 2*hl + {0,1}.
  const float* aptr = &sA[l15][2 * hl];

#pragma unroll 8
  for (int k = 0; k < IN_FEAT; k += 4) {
    v2f a = *(const v2f*)aptr;         // ds_load_b64 (paired by compiler)
    aptr += 4;
#pragma unroll
    for (int nt = 0; nt < 4; ++nt) {
      v2f bb = *(gcv2fp)bptr[nt];      // global_load_b64 (L2-resident weights)
      bptr[nt] += 4;
      acc[nt] = __builtin_amdgcn_wmma_f32_16x16x4_f32(
          false, a, false, bb, (short)0, acc[nt], false, false);
    }
  }

  // C/D layout: lane -> column l15; VGPR r -> row r + 8*hl.
#pragma unroll
  for (int nt = 0; nt < 4; ++nt) {
    const int col = ncol0 + nt * 16 + l15;
#pragma unroll
    for (int r = 0; r < 8; ++r) {
      const int m = r + 8 * hl;
      if (m < mcount)
        ((gfp)out)[(size_t)sTok[m] * OUT_FEAT + col] = acc[nt][r];
    }
  }
}

extern "C" void kernel_launch(void* const* d_in, const int* in_sizes, int n_in,
                              void* d_out, int out_size, void* d_ws, size_t ws_size,
                              hipStream_t stream) {
  (void)in_sizes; (void)n_in; (void)out_size; (void)ws_size;
  const float* inp    = (const float*)d_in[0];
  const int*   gate   = (const int*)d_in[1];
  const float* weight = (const float*)d_in[2];
  float* out = (float*)d_out;
  MoeWs* ws  = (MoeWs*)d_ws;

  moe_zero<<<1, 64, 0, stream>>>(ws);
  moe_hist<<<BATCH / 256, 256, 0, stream>>>(gate, ws);
  moe_scan<<<1, 32, 0, stream>>>(ws);
  moe_scatter<<<BATCH / 256, 256, 0, stream>>>(gate, ws);
  moe_gemm<<<dim3(MAX_TILES, OUT_FEAT / 256), 128, 0, stream>>>(inp, weight, out, ws);
}